// MSDeformAttn_6339371729445
// MI455X (gfx1250) — compile-verified
//
#include <hip/hip_runtime.h>
#include <hip/hip_bf16.h>
#include <math.h>

typedef __attribute__((ext_vector_type(2))) float v2f;
typedef __attribute__((ext_vector_type(4))) float v4f;
typedef __attribute__((ext_vector_type(8))) float v8f;

#define LDS_STRIDE 132   // 132 % 64 == 4 -> conflict-free b64 fragment reads

// ---------------------------------------------------------------------------
// C[MxN] = A[Mx128] @ W[Nx128]^T + bias[N]     (N multiple of 128)
// Block: 256 thr (8 waves) -> 128 rows x 128 cols. Both tiles staged in LDS
// via coalesced b128, WMMA fragments read as conflict-free ds_load_b64 with
// one-k-step software double buffering.
// ---------------------------------------------------------------------------
__global__ __launch_bounds__(256)
void gemm128_nt_bias(const float* __restrict__ A,
                     const float* __restrict__ Wt,   // N x 128, row-major
                     const float* __restrict__ bias,
                     float* __restrict__ C,
                     int M, int N)
{
    __shared__ float smem[256 * LDS_STRIDE];     // [0..127]=W rows, [128..255]=A rows

    const int tid   = threadIdx.x;
    const int wave  = tid >> 5;
    const int lane  = tid & 31;
    const int l16   = lane & 15;
    const int khalf = lane >> 4;                 // 0 or 1
    const int m0blk = blockIdx.x * 128;
    const int m0    = m0blk + wave * 16;
    const int n0    = blockIdx.y * 128;

    // ---- cooperative staging: each thread copies half a row (64 floats) ----
    {
        const int r    = tid >> 1;               // 0..127
        const int half = (tid & 1) * 64;
        const float* wsrc = Wt + (size_t)(n0 + r) * 128 + half;
        float*       wdst = &smem[r * LDS_STRIDE + half];
        const int ar = (m0blk + r < M) ? (m0blk + r) : (M - 1);
        const float* asrc = A + (size_t)ar * 128 + half;
        float*       adst = &smem[(128 + r) * LDS_STRIDE + half];
#pragma unroll
        for (int i = 0; i < 16; ++i) {
            v4f wv = *(const v4f*)(wsrc + i * 4);
            v4f av = *(const v4f*)(asrc + i * 4);
            *(v4f*)(wdst + i * 4) = wv;
            *(v4f*)(adst + i * 4) = av;
        }
    }
    __syncthreads();

    const float* la = &smem[(128 + wave * 16 + l16) * LDS_STRIDE + khalf * 2];
    const float* lb[8];
#pragma unroll
    for (int t = 0; t < 8; ++t)
        lb[t] = &smem[(t * 16 + l16) * LDS_STRIDE + khalf * 2];

    v8f acc[8] = {};

    // ---- K loop: 32 steps of V_WMMA_F32_16X16X4_F32, double-buffered frags ----
    v2f a_c = *(const v2f*)(la);
    v2f b_c[8];
#pragma unroll
    for (int t = 0; t < 8; ++t) b_c[t] = *(const v2f*)(lb[t]);

#pragma unroll
    for (int kk = 0; kk < 32; ++kk) {
        const int kn = (kk < 31) ? (kk + 1) : kk;
        v2f a_n = *(const v2f*)(la + kn * 4);
        v2f b_n[8];
#pragma unroll
        for (int t = 0; t < 8; ++t) b_n[t] = *(const v2f*)(lb[t] + kn * 4);

#pragma unroll
        for (int t = 0; t < 8; ++t)
            acc[t] = __builtin_amdgcn_wmma_f32_16x16x4_f32(
                false, a_c, false, b_n == nullptr ? b_c[t] : b_c[t], (short)0,
                acc[t], false, false);

        a_c = a_n;
#pragma unroll
        for (int t = 0; t < 8; ++t) b_c[t] = b_n[t];
    }

    // ---- epilogue: bias + predicated stores (two 64B segments per VGPR) ----
    const int mtop = m0 + khalf * 8;
#pragma unroll
    for (int t = 0; t < 8; ++t) {
        const int nn = n0 + t * 16 + l16;
        const float bv = bias[nn];
#pragma unroll
        for (int r = 0; r < 8; ++r) {
            const int m = mtop + r;
            if (m < M) C[(size_t)m * N + nn] = acc[t][r] + bv;
        }
    }
}

// ---------------------------------------------------------------------------
// Softmax(16) + multi-scale bilinear gather. One thread per (query-row, head).
// dh = 16 floats = 64 contiguous bytes per corner -> 4x v4f loads.
// ---------------------------------------------------------------------------
__global__ __launch_bounds__(256)
void msdeform_core(const float* __restrict__ ref_points, // (216,153,4,2)
                   const int*   __restrict__ lvl_start,  // (4)
                   const int*   __restrict__ shp,        // (4,2) = [W,H]
                   const float* __restrict__ off_ws,     // (33048,256)
                   const float* __restrict__ aw_ws,      // (33048,128)
                   const float* __restrict__ value_h,    // (216,4080,128)
                   float* __restrict__ out_ws,           // (33048,128)
                   int total)
{
    const int gid = blockIdx.x * blockDim.x + threadIdx.x;
    if (gid >= total) return;
    const int h  = gid & 7;
    const int m  = gid >> 3;        // b2*153 + q
    const int b2 = m / 153;

    // ---- softmax over the 16 (level,point) logits of this head ----
    const float* lg = aw_ws + (size_t)m * 128 + h * 16;
    float w[16];
    float mx = lg[0];
#pragma unroll
    for (int i = 1; i < 16; ++i) mx = fmaxf(mx, lg[i]);
    float s = 0.f;
#pragma unroll
    for (int i = 0; i < 16; ++i) { w[i] = __expf(lg[i] - mx); s += w[i]; }
    const float inv = 1.0f / s;
#pragma unroll
    for (int i = 0; i < 16; ++i) w[i] *= inv;

    const float* offp = off_ws + (size_t)m * 256 + h * 32;
    const float* refp = ref_points + (size_t)m * 8;
    const float* vb   = value_h + (size_t)b2 * 4080 * 128 + h * 16;

    v4f acc0 = {}, acc1 = {}, acc2 = {}, acc3 = {};

    for (int l = 0; l < 4; ++l) {
        const int Wl = shp[l * 2 + 0];
        const int Hl = shp[l * 2 + 1];
        const int st = lvl_start[l];
        const float fW = (float)Wl, fH = (float)Hl;
        const float invW = 1.0f / fW, invH = 1.0f / fH;
        const float rx = refp[l * 2 + 0];
        const float ry = refp[l * 2 + 1];
        const float* vlvl = vb + (size_t)st * 128;

        auto corner = [&](int xi, int yi, float cw) {
            if (xi >= 0 && xi < Wl && yi >= 0 && yi < Hl) {
                const float* vp = vlvl + (size_t)(yi * Wl + xi) * 128;
                acc0 += cw * *(const v4f*)(vp + 0);
                acc1 += cw * *(const v4f*)(vp + 4);
                acc2 += cw * *(const v4f*)(vp + 8);
                acc3 += cw * *(const v4f*)(vp + 12);
            }
        };

#pragma unroll
        for (int p = 0; p < 4; ++p) {
            const float ox = offp[(l * 4 + p) * 2 + 0];
            const float oy = offp[(l * 4 + p) * 2 + 1];
            const float px = (rx + ox * invW) * fW - 0.5f;
            const float py = (ry + oy * invH) * fH - 0.5f;
            const float fx0 = floorf(px), fy0 = floorf(py);
            const int x0 = (int)fx0, y0 = (int)fy0;
            const float fx = px - fx0, fy = py - fy0;
            const float aw = w[l * 4 + p];
            const float w00 = (1.f - fx) * (1.f - fy) * aw;
            const float w10 = fx * (1.f - fy) * aw;
            const float w01 = (1.f - fx) * fy * aw;
            const float w11 = fx * fy * aw;
            corner(x0,     y0,     w00);
            corner(x0 + 1, y0,     w10);
            corner(x0,     y0 + 1, w01);
            corner(x0 + 1, y0 + 1, w11);
        }
    }

    float* op = out_ws + (size_t)m * 128 + h * 16;
    *(v4f*)(op + 0)  = acc0;
    *(v4f*)(op + 4)  = acc1;
    *(v4f*)(op + 8)  = acc2;
    *(v4f*)(op + 12) = acc3;
}

// ---------------------------------------------------------------------------
extern "C" void kernel_launch(void* const* d_in, const int* in_sizes, int n_in,
                              void* d_out, int out_size, void* d_ws, size_t ws_size,
                              hipStream_t stream)
{
    const float* query   = (const float*)d_in[0];
    const float* refpts  = (const float*)d_in[1];
    const float* value   = (const float*)d_in[2];
    const int*   indices = (const int*)d_in[3];
    const int*   shapes  = (const int*)d_in[4];
    const float* W_off   = (const float*)d_in[5];
    const float* b_off   = (const float*)d_in[6];
    const float* W_attn  = (const float*)d_in[7];
    const float* b_attn  = (const float*)d_in[8];
    const float* W_val   = (const float*)d_in[9];
    const float* b_val   = (const float*)d_in[10];
    const float* W_out   = (const float*)d_in[11];
    const float* b_out   = (const float*)d_in[12];

    const int c   = 128;
    const int bt2 = 8 * 27;          // 216
    const int Lq  = 9 * 17;          // 153
    const int Lv  = 4080;
    const int Mq  = bt2 * Lq;        // 33048
    const int Mv  = bt2 * Lv;        // 881280

    float* new_query = (float*)d_out;                        // Mq * 128
    float* new_value = (float*)d_out + (size_t)Mq * c;       // Mv * 128

    float* off_ws = (float*)d_ws;                            // Mq * 256
    float* aw_ws  = off_ws + (size_t)Mq * 256;               // Mq * 128
    float* out_ws = aw_ws  + (size_t)Mq * 128;               // Mq * 128

    dim3 blk(256);
    const int gMq = (Mq + 127) / 128;                        // 259

    // 1) new_value = value @ W_val^T + b_val   (bandwidth-bound big GEMM)
    gemm128_nt_bias<<<dim3(Mv / 128, 1), blk, 0, stream>>>(value, W_val, b_val,
                                                           new_value, Mv, 128);
    // 2) sampling offsets: q @ W_off^T + b_off  -> (Mq, 256)
    gemm128_nt_bias<<<dim3(gMq, 2), blk, 0, stream>>>(query, W_off, b_off,
                                                      off_ws, Mq, 256);
    // 3) attention logits: q @ W_attn^T + b_attn -> (Mq, 128)
    gemm128_nt_bias<<<dim3(gMq, 1), blk, 0, stream>>>(query, W_attn, b_attn,
                                                      aw_ws, Mq, 128);
    // 4) softmax + bilinear gather -> (Mq, 128)
    const int total = Mq * 8;
    msdeform_core<<<(total + 255) / 256, blk, 0, stream>>>(refpts, indices, shapes,
                                                           off_ws, aw_ws, new_value,
                                                           out_ws, total);
    // 5) new_query = out @ W_out^T + b_out
    gemm128_nt_bias<<<dim3(gMq, 1), blk, 0, stream>>>(out_ws, W_out, b_out,
                                                      new_query, Mq, 128);
}